// MultiScaleRoIAlign_43808666419397
// MI455X (gfx1250) — compile-verified
//
#include <hip/hip_runtime.h>

// MultiScaleRoIAlign for MI455X (gfx1250).
// Separable bilinear+avgpool -> two fp32 V_WMMA_F32_16X16X4_F32 GEMM chains per
// RoI; channels are the WMMA N dimension. One workgroup (4 waves) per RoI.
// Gathers use scalar (SGPR) row bases + provably-unsigned 32-bit per-lane
// column offsets so every load lowers to saddr+voffset scale_offset form.

typedef __attribute__((ext_vector_type(2))) float v2f;
typedef __attribute__((ext_vector_type(8))) float v8f;

#define OUTP 7      // pooled output size
#define NSAMP 14    // OUTP * sampling_ratio
#define NJ 28       // 2 corners per sample
#define NCH 256
#define TROW (OUTP * 16)   // 112 floats per jy row of T
#define OFFMASK 0x00FFFFFFu  // offsets < 2^25 elements; known-bits hint

__device__ __forceinline__ void wave_lds_fence() {
    // Wave-private LDS producer->consumer ordering (DS ops are in-order per
    // wave); stops compiler reordering and waits for cross-lane visibility.
    asm volatile("s_wait_dscnt 0x0" ::: "memory");
    __builtin_amdgcn_wave_barrier();
}

__global__ __launch_bounds__(128)
void msroi_wmma_kernel(const float* __restrict__ f0,
                       const float* __restrict__ f1,
                       const float* __restrict__ f2,
                       const float* __restrict__ f3,
                       const float* __restrict__ boxes,
                       float* __restrict__ out)
{
    __shared__ float s_ax[NJ];
    __shared__ float s_ay[NJ];
    __shared__ int   s_X[NJ];
    __shared__ int   s_Y[NJ];
    __shared__ __attribute__((aligned(16))) float s_T[4][NJ * TROW]; // 50 KB
    __shared__ __attribute__((aligned(16))) float s_O[4][16 * 49];   // 12.5 KB

    const int roi  = blockIdx.x;
    const int t    = threadIdx.x;
    const int bidx = roi >> 8;

    const float bx1 = boxes[roi * 4 + 0];
    const float by1 = boxes[roi * 4 + 1];
    const float bx2 = boxes[roi * 4 + 2];
    const float by2 = boxes[roi * 4 + 3];

    // FPN level mapper (uniform across workgroup)
    float area = fmaxf((bx2 - bx1) * (by2 - by1), 1e-12f);
    int lvl = (int)floorf(4.0f + log2f(sqrtf(area) * (1.0f / 224.0f)) + 1e-6f);
    lvl = min(max(lvl, 2), 5) - 2;                 // 0..3
    const int   H     = 200 >> lvl;                // W == H
    const float scale = 0.25f / (float)(1 << lvl);
    const float* feat = (lvl == 0) ? f0 : (lvl == 1) ? f1 : (lvl == 2) ? f2 : f3;
    const int   HW    = H * H;
    const float* fb   = feat + (long)bidx * NCH * HW;   // uniform base

    // ---- per-RoI sample coords & separable weights (threads 0..27) ----
    if (t < NJ) {
        const bool  isY = (t >= NSAMP);
        const int   i   = isY ? (t - NSAMP) : t;   // sample 0..13
        const float c1  = (isY ? by1 : bx1) * scale;
        const float c2  = (isY ? by2 : bx2) * scale;
        const float len = fmaxf(c2 - c1, 1.0f);
        const float bsz = len * (1.0f / OUTP);
        const float off = (float)(i >> 1) + ((i & 1) ? 0.75f : 0.25f);
        float p = c1 + off * bsz;
        const bool valid = (p > -1.0f) && (p < (float)H);
        float pc = fmaxf(p, 0.0f);
        int lo = min((int)floorf(pc), H - 1);
        int hi = min(lo + 1, H - 1);
        if (lo >= H - 1) pc = (float)lo;
        float fr  = pc - (float)lo;
        float wlo = valid ? 0.5f * (1.0f - fr) : 0.0f;   // 0.5 = 1/SR avg factor
        float whi = valid ? 0.5f * fr          : 0.0f;
        if (isY) { s_Y[2 * i] = lo; s_Y[2 * i + 1] = hi;
                   s_ay[2 * i] = wlo; s_ay[2 * i + 1] = whi; }
        else     { s_X[2 * i] = lo; s_X[2 * i + 1] = hi;
                   s_ax[2 * i] = wlo; s_ax[2 * i + 1] = whi; }
    }
    __syncthreads();

    const int wave = t >> 5;
    const int lane = t & 31;
    const int hl   = lane >> 4;   // wave half: K rows 2h / 2h+1
    const int n    = lane & 15;   // N index (channel in tile) == M row id

    float* Tw = &s_T[wave][0];
    float* Ow = &s_O[wave][0];

    // ---- hoisted loop-invariant WMMA A operands & column indices ----
    v2f aX[7], aY[7];
    unsigned xi0[7], xi1[7];
    #pragma unroll
    for (int k = 0; k < 7; ++k) {
        const float ax0 = s_ax[4 * k + 2 * hl];
        const float ax1 = s_ax[4 * k + 2 * hl + 1];
        const float ay0 = s_ay[4 * k + 2 * hl];
        const float ay1 = s_ay[4 * k + 2 * hl + 1];
        const bool  sel = (n == k);
        aX[k].x = sel ? ax0 : 0.0f;  aX[k].y = sel ? ax1 : 0.0f;
        aY[k].x = sel ? ay0 : 0.0f;  aY[k].y = sel ? ay1 : 0.0f;
        xi0[k] = (unsigned)s_X[4 * k + 2 * hl];
        xi1[k] = (unsigned)s_X[4 * k + 2 * hl + 1];
    }
    // Scalarized row offsets: constant-indexed after full unroll -> SGPRs
    int yrow[NJ];
    #pragma unroll
    for (int j = 0; j < NJ; ++j)
        yrow[j] = __builtin_amdgcn_readfirstlane(s_Y[j] * H);

    for (int tile = 0; tile < 4; ++tile) {
        const int      cbase = wave * 64 + tile * 16;
        const unsigned choff = (unsigned)((cbase + n) * HW);   // < 2^25
        unsigned cx0[7], cx1[7];
        #pragma unroll
        for (int k = 0; k < 7; ++k) {
            // Masking is a no-op on the value but pins high bits to zero so
            // the backend can prove the offset unsigned -> saddr+scale_offset.
            cx0[k] = (choff + xi0[k]) & OFFMASK;
            cx1[k] = (choff + xi1[k]) & OFFMASK;
        }

        // ---- stage 1: T[jy, pw, c] = sum_jx Ax * P[jy, jx, c] ----
        #pragma unroll                                 // full unroll: SGPR bases
        for (int jy = 0; jy < NJ; ++jy) {
            const float* __restrict__ rp = fb + yrow[jy];  // scalar base
            float b0[7], b1[7];
            #pragma unroll
            for (int k = 0; k < 7; ++k) {          // 14-gather clause
                b0[k] = rp[cx0[k]];
                b1[k] = rp[cx1[k]];
            }
            v8f acc = {};
            #pragma unroll
            for (int k = 0; k < 7; ++k) {
                v2f bm; bm.x = b0[k]; bm.y = b1[k];
                acc = __builtin_amdgcn_wmma_f32_16x16x4_f32(
                        false, aX[k], false, bm, (short)0, acc, false, false);
            }
            if (hl == 0) {                         // rows M=0..6 in lanes 0-15
                #pragma unroll
                for (int r = 0; r < OUTP; ++r)
                    Tw[jy * TROW + r * 16 + n] = acc[r];
            }
        }
        wave_lds_fence();

        // ---- stage 2: out[ph, pw, c] = sum_jy Ay * T[jy, pw, c] ----
        const float* __restrict__ T2 = Tw + 2 * hl * TROW;
        #pragma unroll                                 // immediate ds offsets
        for (int pw = 0; pw < OUTP; ++pw) {
            const float* __restrict__ tp = T2 + pw * 16 + n;
            float c0[7], c1[7];
            #pragma unroll
            for (int k = 0; k < 7; ++k) {
                c0[k] = tp[4 * k * TROW];
                c1[k] = tp[(4 * k + 1) * TROW];
            }
            v8f acc = {};
            #pragma unroll
            for (int k = 0; k < 7; ++k) {
                v2f bm; bm.x = c0[k]; bm.y = c1[k];
                acc = __builtin_amdgcn_wmma_f32_16x16x4_f32(
                        false, aY[k], false, bm, (short)0, acc, false, false);
            }
            if (hl == 0) {
                #pragma unroll
                for (int r = 0; r < OUTP; ++r)
                    Ow[n * 49 + r * 7 + pw] = acc[r];   // [c][ph*7+pw]
            }
        }
        wave_lds_fence();

        // ---- coalesced b128 output store: contiguous 16*49 floats ----
        float* obase = out + ((long)roi * NCH + cbase) * 49;
        #pragma unroll
        for (int i0 = 0; i0 < 16 * 49; i0 += 128) {
            const int i = i0 + lane * 4;
            if (i < 16 * 49) {
                float4 v = *(const float4*)&Ow[i];
                *(float4*)&obase[i] = v;
            }
        }
        wave_lds_fence();   // protect Tw/Ow reuse next tile
    }
}

extern "C" void kernel_launch(void* const* d_in, const int* in_sizes, int n_in,
                              void* d_out, int out_size, void* d_ws, size_t ws_size,
                              hipStream_t stream) {
    const float* f0    = (const float*)d_in[0];
    const float* f1    = (const float*)d_in[1];
    const float* f2    = (const float*)d_in[2];
    const float* f3    = (const float*)d_in[3];
    const float* boxes = (const float*)d_in[4];
    float* out = (float*)d_out;

    msroi_wmma_kernel<<<dim3(512), dim3(128), 0, stream>>>(f0, f1, f2, f3, boxes, out);
}